// MLP_57277683860079
// MI455X (gfx1250) — compile-verified
//
#include <hip/hip_runtime.h>

typedef float v2f __attribute__((ext_vector_type(2)));
typedef float v8f __attribute__((ext_vector_type(8)));

#define N_NODES 50000
#define IN_CH   128
#define HIDC    64
#define OUTC    64
#define EPSV    1e-5f
#define NSLOPE  0.01f
#define MTILES  (N_NODES / 16)   // 3125 exact

// monotone float -> u32 key (order-preserving), so min() can use u32 atomics
__device__ __forceinline__ unsigned fkey(float f) {
  unsigned b = __float_as_uint(f);
  return (b & 0x80000000u) ? ~b : (b | 0x80000000u);
}
// inverse; keys >= key(+inf)=0xFF800000 mean "no edge touched this slot" -> 0.0 (PyG fill)
__device__ __forceinline__ float funkey_or_zero(unsigned k) {
  if (k >= 0xFF800000u) return 0.0f;
  unsigned b = (k & 0x80000000u) ? (k & 0x7FFFFFFFu) : ~k;
  return __uint_as_float(b);
}

// ---------------- Kernel 1: h = LayerNorm(LeakyReLU(x @ W1^T + b1)) ----------------
// One wave computes a 16x64 tile using v_wmma_f32_16x16x4_f32.
__global__ __launch_bounds__(128) void k1_gemm_ln(
    const float* __restrict__ x, const float* __restrict__ W1,
    const float* __restrict__ b1, const float* __restrict__ gamma,
    const float* __restrict__ beta, float* __restrict__ h)
{
  const int tile = blockIdx.x * 4 + (threadIdx.x >> 5);
  if (tile >= MTILES) return;                 // whole-wave exit: EXEC stays all-1s for WMMA
  const int lane = threadIdx.x & 31;
  const int half = lane >> 4;                 // A/B: lanes 16-31 hold K+2
  const int l15  = lane & 15;
  const int m0   = tile * 16;

  v8f acc[4] = {};
  const float* arow = x + (size_t)(m0 + l15) * IN_CH + half * 2;

  for (int k0 = 0; k0 < IN_CH; k0 += 4) {
    float2 av = *(const float2*)(arow + k0);          // A[m][k0+half*2 .. +1]
    v2f a; a.x = av.x; a.y = av.y;
#pragma unroll
    for (int t = 0; t < 4; ++t) {
      // B[k][n] = W1[n][k] (W1 is [64,128] row-major): lane holds col t*16+l15
      const float* bp = W1 + (size_t)(t * 16 + l15) * IN_CH + k0 + half * 2;
      float2 bv = *(const float2*)bp;
      v2f b; b.x = bv.x; b.y = bv.y;
      acc[t] = __builtin_amdgcn_wmma_f32_16x16x4_f32(
          false, a, false, b, (short)0, acc[t], false, false);
    }
  }

  float bcol[4], gcol[4], zcol[4];
#pragma unroll
  for (int t = 0; t < 4; ++t) {
    const int col = t * 16 + l15;
    bcol[t] = b1[col]; gcol[t] = gamma[col]; zcol[t] = beta[col];
  }

  float vals[4][8];
#pragma unroll
  for (int t = 0; t < 4; ++t)
#pragma unroll
    for (int r = 0; r < 8; ++r) {
      float v = acc[t][r] + bcol[t];
      vals[t][r] = (v > 0.0f) ? v : NSLOPE * v;       // LeakyReLU
    }

  // LayerNorm per row: row (m0 + half*8 + r) lives in one 16-lane half (C/D layout)
#pragma unroll
  for (int r = 0; r < 8; ++r) {
    float s = 0.0f, q = 0.0f;
#pragma unroll
    for (int t = 0; t < 4; ++t) { s += vals[t][r]; q += vals[t][r] * vals[t][r]; }
#pragma unroll
    for (int m = 1; m < 16; m <<= 1) {                // stays within each 16-lane half
      s += __shfl_xor(s, m, 32);
      q += __shfl_xor(q, m, 32);
    }
    const float mu  = s * (1.0f / HIDC);
    const float var = q * (1.0f / HIDC) - mu * mu;
    const float rs  = rsqrtf(var + EPSV);
    const int row = m0 + half * 8 + r;
#pragma unroll
    for (int t = 0; t < 4; ++t) {
      const int col = t * 16 + l15;
      h[(size_t)row * HIDC + col] = (vals[t][r] - mu) * rs * gcol[t] + zcol[t];
    }
  }
}

// ---------------- Kernel 2: scatter segment-min over edges ----------------
// One thread per (edge, channel); single hardware global_atomic_min_u32 per element.
__global__ __launch_bounds__(256) void k2_scatter_min(
    const int* __restrict__ ei, const float* __restrict__ h,
    unsigned* __restrict__ aggkey, int E)
{
  const int tid = blockIdx.x * 256 + threadIdx.x;
  const int e = tid >> 6;
  if (e >= E) return;
  const int c   = tid & 63;
  const int src = ei[e];          // edge_index[0][e]
  const int dst = ei[E + e];      // edge_index[1][e]
  const float v = h[(size_t)src * HIDC + c];     // coalesced 256B per edge
  atomicMin(aggkey + (size_t)dst * HIDC + c, fkey(v));
}

// ---------------- Kernel 3: out = decode(agg) @ W2^T + b2 ----------------
__global__ __launch_bounds__(128) void k3_gemm2(
    const unsigned* __restrict__ aggkey, const float* __restrict__ W2,
    const float* __restrict__ b2, float* __restrict__ out)
{
  const int tile = blockIdx.x * 4 + (threadIdx.x >> 5);
  if (tile >= MTILES) return;
  const int lane = threadIdx.x & 31;
  const int half = lane >> 4;
  const int l15  = lane & 15;
  const int m0   = tile * 16;

  v8f acc[4] = {};
  const unsigned* arow = aggkey + (size_t)(m0 + l15) * HIDC + half * 2;

  for (int k0 = 0; k0 < HIDC; k0 += 4) {
    uint2 ak = *(const uint2*)(arow + k0);
    v2f a; a.x = funkey_or_zero(ak.x); a.y = funkey_or_zero(ak.y);
#pragma unroll
    for (int t = 0; t < 4; ++t) {
      const float* bp = W2 + (size_t)(t * 16 + l15) * HIDC + k0 + half * 2;
      float2 bv = *(const float2*)bp;
      v2f b; b.x = bv.x; b.y = bv.y;
      acc[t] = __builtin_amdgcn_wmma_f32_16x16x4_f32(
          false, a, false, b, (short)0, acc[t], false, false);
    }
  }

#pragma unroll
  for (int t = 0; t < 4; ++t) {
    const int col = t * 16 + l15;
    const float bb = b2[col];
#pragma unroll
    for (int r = 0; r < 8; ++r) {
      const int row = m0 + half * 8 + r;
      out[(size_t)row * OUTC + col] = acc[t][r] + bb;
    }
  }
}

extern "C" void kernel_launch(void* const* d_in, const int* in_sizes, int n_in,
                              void* d_out, int out_size, void* d_ws, size_t ws_size,
                              hipStream_t stream) {
  const float* x     = (const float*)d_in[0];
  // d_in[1] = x_e (unused by the module)
  const int*   ei    = (const int*)d_in[2];
  const float* W1    = (const float*)d_in[3];
  const float* b1    = (const float*)d_in[4];
  const float* gamma = (const float*)d_in[5];
  const float* beta  = (const float*)d_in[6];
  const float* W2    = (const float*)d_in[7];
  const float* b2    = (const float*)d_in[8];
  float* out = (float*)d_out;
  const int E = in_sizes[2] / 2;

  // workspace: h [N,64] f32, then aggkey [N,64] u32  (25.6 MB total)
  float*    h      = (float*)d_ws;
  unsigned* aggkey = (unsigned*)((char*)d_ws + (size_t)N_NODES * HIDC * sizeof(float));

  hipMemsetAsync(aggkey, 0xFF, (size_t)N_NODES * HIDC * sizeof(unsigned), stream);

  {
    dim3 blk(128), grd((MTILES + 3) / 4);
    k1_gemm_ln<<<grd, blk, 0, stream>>>(x, W1, b1, gamma, beta, h);
  }
  {
    const long long total = (long long)E * 64;
    dim3 blk(256), grd((unsigned)((total + 255) / 256));
    k2_scatter_min<<<grd, blk, 0, stream>>>(ei, h, aggkey, E);
  }
  {
    dim3 blk(128), grd((MTILES + 3) / 4);
    k3_gemm2<<<grd, blk, 0, stream>>>(aggkey, W2, b2, out);
  }
}